// DenseEmbed_42906723287515
// MI455X (gfx1250) — compile-verified
//
#include <hip/hip_runtime.h>

// T=8, B=64, D=512, E=256  ->  TB = T*B = 512 rows of (D x E) broadcast product.
// out[(tb*D + i)*E + e] = x[tb*D + i] * W[i*E + e] + b[e]
//
// Store-bandwidth bound: 268 MB f32 output @ 23.3 TB/s ~= 11.5 us floor.
// Strategy: async-stage W tile (CDNA5 global_load_async_to_lds, ASYNCcnt) into
// LDS, stream output with non-temporal 128-bit stores.

typedef float vf4 __attribute__((ext_vector_type(4)));
typedef int   vi4 __attribute__((ext_vector_type(4)));

#define AS_GLOBAL __attribute__((address_space(1)))
#define AS_LOCAL  __attribute__((address_space(3)))

__device__ __forceinline__ void async_copy_b128(const void* gsrc, void* lds_dst) {
#if __has_builtin(__builtin_amdgcn_global_load_async_to_lds_b128)
  __builtin_amdgcn_global_load_async_to_lds_b128(
      (AS_GLOBAL vi4*)gsrc, (AS_LOCAL vi4*)lds_dst, /*offset=*/0, /*cpol=*/0);
#else
  *(vf4*)lds_dst = *(const vf4*)gsrc;  // synchronous fallback
#endif
}

__device__ __forceinline__ void async_copy_b32(const void* gsrc, void* lds_dst) {
#if __has_builtin(__builtin_amdgcn_global_load_async_to_lds_b32)
  __builtin_amdgcn_global_load_async_to_lds_b32(
      (AS_GLOBAL int*)gsrc, (AS_LOCAL int*)lds_dst, /*offset=*/0, /*cpol=*/0);
#else
  *(float*)lds_dst = *(const float*)gsrc;  // synchronous fallback
#endif
}

__device__ __forceinline__ void wait_async_zero() {
#if __has_builtin(__builtin_amdgcn_s_wait_asynccnt)
  __builtin_amdgcn_s_wait_asynccnt(0);
#else
  asm volatile("s_wait_asynccnt 0" ::: "memory");
#endif
}

__global__ __launch_bounds__(256) void DenseEmbed_bcast_kernel(
    const float* __restrict__ x,   // (TB=512, D=512)
    const float* __restrict__ W,   // (D=512, E=256)
    const float* __restrict__ b,   // (E=256)
    float* __restrict__ out)       // (TB, D, E)
{
  constexpr int D = 512;
  constexpr int E4 = 64;        // E/4 = 64 vf4 per row

  __shared__ vf4   ws[16 * E4]; // W tile: [i_local][e4]  = 16 KB
  __shared__ float xs[16 * 16]; // x tile: [tb_local][i_local] = 1 KB

  const int tid     = threadIdx.x;
  const int i_base  = (blockIdx.x & 31) << 4;  // 32 i-tiles of 16
  const int tb_base = (blockIdx.x >> 5) << 4;  // 32 tb-tiles of 16

  // ---- stage W tile: 16 rows x 64 vf4 = 1024 vf4, 4 per thread (async DMA) ----
  {
    const vf4* wg = (const vf4*)W + (size_t)i_base * E4;
#pragma unroll
    for (int k = 0; k < 4; ++k) {
      const int idx = tid + (k << 8);
      async_copy_b128(wg + idx, &ws[idx]);
    }
  }
  // ---- stage x tile: 256 floats, 1 per thread (async DMA) ----
  {
    const int tb_l = tid >> 4;
    const int i_l  = tid & 15;
    async_copy_b32(x + (size_t)(tb_base + tb_l) * D + i_base + i_l, &xs[tid]);
  }
  wait_async_zero();   // per-wave ASYNCcnt drain
  __syncthreads();     // make LDS writes visible to all 8 waves

  // ---- streaming compute: each 64-lane group owns one 1 KB output row/iter ----
  const int lane_e = tid & 63;  // e4 index 0..63
  const int rowsel = tid >> 6;  // which of 4 rows this iteration
  const vf4 b4 = *((const vf4*)b + lane_e);

#pragma unroll 4
  for (int r = 0; r < 64; ++r) {
    const int row  = (r << 2) + rowsel;  // 0..255 = tb_l*16 + i_l
    const int tb_l = row >> 4;
    const int i_l  = row & 15;

    const float xv = xs[row];                  // LDS broadcast within 64 lanes
    const vf4   w4 = ws[i_l * E4 + lane_e];    // ds_load_b128
    const vf4   o  = xv * w4 + b4;             // 2x v_pk_fma_f32

    const size_t off = ((size_t)(tb_base + tb_l) * D + (i_base + i_l)) * E4 + lane_e;
    __builtin_nontemporal_store(o, (vf4*)out + off);  // NT global_store_b128
  }
}

extern "C" void kernel_launch(void* const* d_in, const int* in_sizes, int n_in,
                              void* d_out, int out_size, void* d_ws, size_t ws_size,
                              hipStream_t stream) {
  const float* x = (const float*)d_in[0];  // (8,64,512) f32
  const float* W = (const float*)d_in[1];  // (512,256) f32
  const float* b = (const float*)d_in[2];  // (256,) f32
  float* out = (float*)d_out;              // (8,64,512,256) f32

  (void)in_sizes; (void)n_in; (void)out_size; (void)d_ws; (void)ws_size;

  // 32 i-tiles x 32 tb-tiles = 1024 blocks, 256 threads (8 wave32s) each.
  DenseEmbed_bcast_kernel<<<dim3(1024), dim3(256), 0, stream>>>(x, W, b, out);
}